// MaximumLikelihoodDetector_55018531062638
// MI455X (gfx1250) — compile-verified
//
#include <hip/hip_runtime.h>
#include <math.h>

// MI455X / gfx1250, wave32.
// Maximum-likelihood QAM16 detector, restructured as:
//   exponents(V x B) = Features(V x 28) @ Coef^T(28 x B)   via V_WMMA_F32_16X16X4_F32
// with fused bitwise-subset logsumexp (no 16MB intermediate, no gathers).
// Round 2: hardware FP atomics (ds_add_f32 / global_atomic_add_f32) via
// unsafeAtomicAdd instead of CAS-loop atomicAdd on the hot accumulators.

typedef __attribute__((ext_vector_type(2))) float v2f;
typedef __attribute__((ext_vector_type(8))) float v8f;

#define NB 64
#define NM 8
#define NK 4
#define NVTOT 65536

__device__ __forceinline__ float round_qam_level(float t) {
    // nearest of {-3,-1,1,3}/sqrt(10)
    float a = t * 3.1622776601683795f;
    float o = 2.0f * rintf(0.5f * (a - 1.0f)) + 1.0f;
    o = fminf(3.0f, fmaxf(-3.0f, o));
    return o * 0.31622776601683794f;
}

// ---------------------------------------------------------------------------
// Kernel A: per-batch complex Cholesky whitening + Gram/matched-filter ->
// 32-entry coefficient vector per batch (feature-space dual of the exponent).
// ---------------------------------------------------------------------------
__global__ __launch_bounds__(64)
void mld_prep_kernel(const float* __restrict__ yr, const float* __restrict__ yi,
                     const float* __restrict__ hr, const float* __restrict__ hi,
                     const float* __restrict__ sr, const float* __restrict__ si,
                     float* __restrict__ coef) {
    int b = threadIdx.x;
    if (b >= NB) return;

    float Lr[NM][NM], Li[NM][NM];
    for (int i = 0; i < NM; ++i)
        for (int j = 0; j < NM; ++j) {
            Lr[i][j] = sr[b * NM * NM + i * NM + j];
            Li[i][j] = si[b * NM * NM + i * NM + j];
        }

    // Complex Cholesky: S = L L^H (lower). Whitening by L^{-1} gives exponents
    // identical to the reference's Hermitian S^{-1/2} (same S^{-1} quadratic form).
    for (int j = 0; j < NM; ++j) {
        float d = Lr[j][j];
        for (int p = 0; p < j; ++p) d -= Lr[j][p] * Lr[j][p] + Li[j][p] * Li[j][p];
        d = fmaxf(d, 1e-20f);
        float lj = sqrtf(d);
        float inv = 1.0f / lj;
        Lr[j][j] = lj; Li[j][j] = 0.0f;
        for (int i = j + 1; i < NM; ++i) {
            float ar = Lr[i][j], ai = Li[i][j];
            for (int p = 0; p < j; ++p) {
                // a -= L[i][p] * conj(L[j][p])
                ar -= Lr[i][p] * Lr[j][p] + Li[i][p] * Li[j][p];
                ai -= Li[i][p] * Lr[j][p] - Lr[i][p] * Li[j][p];
            }
            Lr[i][j] = ar * inv; Li[i][j] = ai * inv;
        }
    }

    // Forward solve L W = [y | H]  (5 RHS columns)
    float Wr[NM][5], Wi[NM][5];
    for (int i = 0; i < NM; ++i) {
        for (int c = 0; c < 5; ++c) {
            float ar = (c == 0) ? yr[b * NM + i] : hr[(b * NM + i) * NK + (c - 1)];
            float ai = (c == 0) ? yi[b * NM + i] : hi[(b * NM + i) * NK + (c - 1)];
            for (int p = 0; p < i; ++p) {
                ar -= Lr[i][p] * Wr[p][c] - Li[i][p] * Wi[p][c];
                ai -= Lr[i][p] * Wi[p][c] + Li[i][p] * Wr[p][c];
            }
            float inv = 1.0f / Lr[i][i];
            Wr[i][c] = ar * inv; Wi[i][c] = ai * inv;
        }
    }

    // Gram G = H~^H H~ (4x4 Hermitian), z = H~^H y~
    float Gr[NK][NK], Gi[NK][NK], zr[NK], zi[NK];
    for (int j = 0; j < NK; ++j)
        for (int k = 0; k < NK; ++k) {
            float gr = 0.0f, gi = 0.0f;
            for (int m = 0; m < NM; ++m) {
                gr += Wr[m][j + 1] * Wr[m][k + 1] + Wi[m][j + 1] * Wi[m][k + 1];
                gi += Wr[m][j + 1] * Wi[m][k + 1] - Wi[m][j + 1] * Wr[m][k + 1];
            }
            Gr[j][k] = gr; Gi[j][k] = gi;
        }
    for (int k = 0; k < NK; ++k) {
        float ar = 0.0f, ai = 0.0f;
        for (int m = 0; m < NM; ++m) {
            ar += Wr[m][k + 1] * Wr[m][0] + Wi[m][k + 1] * Wi[m][0];
            ai += Wr[m][k + 1] * Wi[m][0] - Wi[m][k + 1] * Wr[m][0];
        }
        zr[k] = ar; zi[k] = ai;
    }

    // Matched-filter hard decision -> exponent shift so max exponent ~ 0
    // (keeps exp() in comfortable f32 range; shift cancels in LLR difference)
    const int PJ[6] = {0, 0, 0, 1, 1, 2};
    const int PK[6] = {1, 2, 3, 2, 3, 3};
    float xhr[NK], xhi[NK];
    for (int k = 0; k < NK; ++k) {
        float inv = 1.0f / Gr[k][k];
        xhr[k] = round_qam_level(zr[k] * inv);
        xhi[k] = round_qam_level(zi[k] * inv);
    }
    float rzx = 0.0f, xgx = 0.0f;
    for (int k = 0; k < NK; ++k) {
        rzx += zr[k] * xhr[k] + zi[k] * xhi[k];
        xgx += Gr[k][k] * (xhr[k] * xhr[k] + xhi[k] * xhi[k]);
    }
    for (int t = 0; t < 6; ++t) {
        int j = PJ[t], k = PK[t];
        float prh = xhr[j] * xhr[k] + xhi[j] * xhi[k];
        float pih = xhr[j] * xhi[k] - xhi[j] * xhr[k];
        xgx += 2.0f * (Gr[j][k] * prh - Gi[j][k] * pih);
    }
    float cc = -(2.0f * rzx - xgx); // = -||y~||^2 + MF-residual shift

    // Coefficient vector: dot(feat_v, coef_b) = shifted exponent e'(b,v)
    float* cb = coef + b * 32;
    for (int k = 0; k < NK; ++k) {
        cb[0 + k]  = 2.0f * zr[k];   // * xr_k
        cb[4 + k]  = 2.0f * zi[k];   // * xi_k
        cb[8 + k]  = -Gr[k][k];      // * |x_k|^2
    }
    for (int t = 0; t < 6; ++t) {
        cb[12 + t] = -2.0f * Gr[PJ[t]][PK[t]];  // * Re(conj(x_j) x_k)
        cb[18 + t] =  2.0f * Gi[PJ[t]][PK[t]];  // * Im(conj(x_j) x_k)
    }
    cb[24] = cc;                     // * 1
    for (int f = 25; f < 32; ++f) cb[f] = 0.0f;
}

// ---------------------------------------------------------------------------
// Kernel B: the V-sweep. 512 blocks x 256 threads (8 waves). Each wave owns a
// 16-candidate slab: builds the 16x28 feature fragment in registers, runs
// 4 N-tiles x 7 K-steps of V_WMMA_F32_16X16X4_F32 against the coef matrix,
// then feeds exp() into 32 bit-subset accumulators per batch in LDS
// (native ds_add_f32), finally flushed with global_atomic_add_f32.
// ---------------------------------------------------------------------------
__global__ __launch_bounds__(256)
void mld_sweep_kernel(const float* __restrict__ coef,
                      const float* __restrict__ pts_r, const float* __restrict__ pts_i,
                      float* __restrict__ gsums) {
    __shared__ float s_coef[NB][34];   // [batch][feature], padded
    __shared__ float s_sums[NB][32];   // [batch][2*(4k+i)+bit]
    __shared__ float s_pr[16], s_pi[16];

    int tid = threadIdx.x;
    for (int t = tid; t < NB * 32; t += 256) s_coef[t >> 5][t & 31] = coef[t];
    for (int t = tid; t < NB * 32; t += 256) s_sums[t >> 5][t & 31] = 0.0f;
    if (tid < 16) { s_pr[tid] = pts_r[tid]; s_pi[tid] = pts_i[tid]; }
    __syncthreads();

    int wave = tid >> 5;
    int lane = tid & 31;
    int m    = lane & 15;
    int h    = lane >> 4;        // lane half: selects K-pair and D-row offset
    bool hiL = (h != 0);
    int v0   = blockIdx.x * 128 + wave * 16;   // 16-aligned slab base
    int v    = v0 + m;                          // this lane's candidate row

    // Decode candidate v -> per-stream constellation points and products
    float xr[4], xi[4], nn[4], qr[6], qi[6];
#pragma unroll
    for (int k = 0; k < 4; ++k) {
        int id = (v >> (4 * (3 - k))) & 15;
        xr[k] = s_pr[id]; xi[k] = s_pi[id];
        nn[k] = xr[k] * xr[k] + xi[k] * xi[k];
    }
    const int PJ[6] = {0, 0, 0, 1, 1, 2};
    const int PK[6] = {1, 2, 3, 2, 3, 3};
#pragma unroll
    for (int t = 0; t < 6; ++t) {
        int j = PJ[t], k = PK[t];
        qr[t] = xr[j] * xr[k] + xi[j] * xi[k];
        qi[t] = xr[j] * xi[k] - xi[j] * xr[k];
    }

    // A fragments for 7 K-steps of 4 (features 0..27), per ISA 16x4 f32 A layout:
    // lanes 0-15 carry K = 4s+0,4s+1 ; lanes 16-31 carry K = 4s+2,4s+3
    v2f a[7];
    a[0].x = hiL ? xr[2] : xr[0];  a[0].y = hiL ? xr[3] : xr[1];
    a[1].x = hiL ? xi[2] : xi[0];  a[1].y = hiL ? xi[3] : xi[1];
    a[2].x = hiL ? nn[2] : nn[0];  a[2].y = hiL ? nn[3] : nn[1];
    a[3].x = hiL ? qr[2] : qr[0];  a[3].y = hiL ? qr[3] : qr[1];
    a[4].x = hiL ? qi[0] : qr[4];  a[4].y = hiL ? qi[1] : qr[5];
    a[5].x = hiL ? qi[4] : qi[2];  a[5].y = hiL ? qi[5] : qi[3];
    a[6].x = hiL ? 0.0f  : 1.0f;   a[6].y = 0.0f;

    int off = hiL ? 2 : 0;

#pragma unroll
    for (int nt = 0; nt < 4; ++nt) {
        int bcol = nt * 16 + m;     // batch column this lane owns in C/D layout
        v8f acc = {};
#pragma unroll
        for (int s = 0; s < 7; ++s) {
            v2f bf;
            bf.x = s_coef[bcol][4 * s + off];
            bf.y = s_coef[bcol][4 * s + off + 1];
            acc = __builtin_amdgcn_wmma_f32_16x16x4_f32(
                false, a[s], false, bf, (short)0, acc, false, false);
        }

        // acc[r] = shifted exponent for batch bcol, candidate v = v0 + 8*h + r
        float w[8];
#pragma unroll
        for (int r = 0; r < 8; ++r) w[r] = __expf(acc[r]);
        float W  = ((w[0] + w[1]) + (w[2] + w[3])) + ((w[4] + w[5]) + (w[6] + w[7]));
        float S0 = (w[1] + w[3]) + (w[5] + w[7]);   // v-bit 0 set (r bit 0)
        float S1 = (w[2] + w[3]) + (w[6] + w[7]);   // v-bit 1 set
        float S2 = (w[4] + w[5]) + (w[6] + w[7]);   // v-bit 2 set

        // subset slot = 2*(15 - p) + bit_p(v);  p = v-bit position
        unsafeAtomicAdd(&s_sums[bcol][31], S0); unsafeAtomicAdd(&s_sums[bcol][30], W - S0);
        unsafeAtomicAdd(&s_sums[bcol][29], S1); unsafeAtomicAdd(&s_sums[bcol][28], W - S1);
        unsafeAtomicAdd(&s_sums[bcol][27], S2); unsafeAtomicAdd(&s_sums[bcol][26], W - S2);

        int vb = v0 + 8 * h;        // bits >= 3 are common to all 8 values
#pragma unroll
        for (int p = 3; p < 16; ++p) {
            int bit = (vb >> p) & 1;
            unsafeAtomicAdd(&s_sums[bcol][2 * (15 - p) + bit], W);
        }
    }

    __syncthreads();
    for (int t = tid; t < NB * 32; t += 256)
        unsafeAtomicAdd(&gsums[t], s_sums[t >> 5][t & 31]);
}

// ---------------------------------------------------------------------------
// Kernel C: LLR = log(sum_bit1) - log(sum_bit0). Shift cancels exactly.
// ---------------------------------------------------------------------------
__global__ __launch_bounds__(256)
void mld_llr_kernel(const float* __restrict__ gsums, float* __restrict__ out) {
    int t = blockIdx.x * blockDim.x + threadIdx.x;
    if (t < NB * 16) {
        int b = t >> 4, j = t & 15;   // j = 4k + i  -> out[b,k,i]
        out[t] = logf(gsums[b * 32 + 2 * j + 1]) - logf(gsums[b * 32 + 2 * j]);
    }
}

extern "C" void kernel_launch(void* const* d_in, const int* in_sizes, int n_in,
                              void* d_out, int out_size, void* d_ws, size_t ws_size,
                              hipStream_t stream) {
    const float* yr = (const float*)d_in[0];
    const float* yi = (const float*)d_in[1];
    const float* hr = (const float*)d_in[2];
    const float* hi = (const float*)d_in[3];
    const float* sr = (const float*)d_in[4];
    const float* si = (const float*)d_in[5];
    const float* pr = (const float*)d_in[6];
    const float* pi = (const float*)d_in[7];
    // d_in[8] (vecs_ind) and d_in[9] (c_idx) are implicit in the v-bit decode.

    float* coef  = (float*)d_ws;                        // 64*32 f32 = 8 KB
    float* gsums = (float*)((char*)d_ws + NB * 32 * 4); // 64*32 f32 = 8 KB

    hipMemsetAsync(gsums, 0, NB * 32 * sizeof(float), stream);
    mld_prep_kernel<<<1, 64, 0, stream>>>(yr, yi, hr, hi, sr, si, coef);
    mld_sweep_kernel<<<NVTOT / 128, 256, 0, stream>>>(coef, pr, pi, gsums);
    mld_llr_kernel<<<(NB * 16 + 255) / 256, 256, 0, stream>>>(gsums, (float*)d_out);
}